// Encoder_13657996001662
// MI455X (gfx1250) — compile-verified
//
#include <hip/hip_runtime.h>
#include <hip/hip_bf16.h>

// ---------------------------------------------------------------------------
// GNN encoder (GIN x3 + BatchNorm + residual VQ) for MI455X / gfx1250.
// fp32 WMMA (V_WMMA_F32_16X16X4_F32) for the MLP GEMMs: the workload is
// memory-bound (edge scatter dominates), so fp32 matrix math is free and
// matches the fp32 reference exactly.
// ---------------------------------------------------------------------------

typedef float v2f __attribute__((ext_vector_type(2)));
typedef float v8f __attribute__((ext_vector_type(8)));

#define DIMF 128
#define NNODES 100000
#define NEDGES 1600000
#define NGRAPHS 512
#define NLAYERS 3
#define NRES 3
#define NCODE 16

static inline int divUp(long long a, long long b) { return (int)((a + b - 1) / b); }

// ------------------------------- utility -----------------------------------

__global__ void zero_kernel(float* __restrict__ p, int n) {
    const int i = blockIdx.x * blockDim.x + threadIdx.x;
    if (i < n) p[i] = 0.0f;
}

__global__ void copy4_kernel(const float4* __restrict__ a, float4* __restrict__ b, int n4) {
    const int i = blockIdx.x * blockDim.x + threadIdx.x;
    if (i < n4) b[i] = a[i];
}

// --------------------------- edge scatter-add ------------------------------
// agg[dst] += h[src]; one 32-lane wave per edge, float4 per lane (128 floats).
__global__ __launch_bounds__(256) void scatter_add_kernel(
    const float* __restrict__ h, const int* __restrict__ src,
    const int* __restrict__ dst, float* __restrict__ agg, int nEdges)
{
    const int e = blockIdx.x * 8 + (threadIdx.x >> 5);
    if (e >= nEdges) return;
    const int lane = threadIdx.x & 31;
    const int s = src[e];
    const int d = dst[e];
    const float4 v = *(const float4*)(h + (size_t)s * DIMF + lane * 4);
    float* o = agg + (size_t)d * DIMF + lane * 4;
    atomicAdd(o + 0, v.x);
    atomicAdd(o + 1, v.y);
    atomicAdd(o + 2, v.z);
    atomicAdd(o + 3, v.w);
}

// ------------------------- fused GIN MLP (WMMA) ----------------------------
// One block = 128 threads (4 waves) = one 16-node row tile.
// GEMM1: Y = relu(X @ W1 + b1) staged in LDS; GEMM2: H = relu(Y @ W2 + b2).
// Each wave computes two 16x16 output tiles per GEMM, K=128 in steps of 4
// via V_WMMA_F32_16X16X4_F32. BatchNorm partial sums folded in at the end.
//
// fp32 WMMA register layouts (wave32):
//   A 16x4 : lanes 0-15 rows 0-15, vgpr0={K0,K2 by half}, vgpr1={K1,K3}
//            -> a.x = A[lane&15][k + 2*(lane>>4)], a.y = same k+1
//   B 4x16 : b.x = B[k + 2*(lane>>4)][lane&15],  b.y = same k+1
//   C/D    : c[r] = D[r + 8*(lane>>4)][lane&15]
__global__ __launch_bounds__(128) void gin_mlp_kernel(
    const float* __restrict__ xin,
    const float* __restrict__ W1, const float* __restrict__ b1,
    const float* __restrict__ W2, const float* __restrict__ b2,
    float* __restrict__ hout,
    float* __restrict__ bn_sum, float* __restrict__ bn_sumsq)
{
    __shared__ float sA[16][DIMF + 4];
    __shared__ float sB[16][DIMF + 4];

    const int tid  = threadIdx.x;
    const int wave = tid >> 5;
    const int lane = tid & 31;
    const int half = lane >> 4;   // 0 or 1
    const int l16  = lane & 15;
    const size_t row0 = (size_t)blockIdx.x * 16;

    // cooperative load of the 16x128 input tile
    for (int i = tid; i < 16 * (DIMF / 4); i += 128) {
        const int m  = i >> 5;          // 32 float4 per row
        const int c4 = (i & 31) << 2;
        const float4 v = *(const float4*)(xin + (row0 + m) * DIMF + c4);
        sA[m][c4 + 0] = v.x; sA[m][c4 + 1] = v.y;
        sA[m][c4 + 2] = v.z; sA[m][c4 + 3] = v.w;
    }
    __syncthreads();

    // ---- GEMM1 + bias + relu -> sB ----
    for (int t = 0; t < 2; ++t) {
        const int ct = wave * 32 + t * 16;
        v8f c = {};
        for (int kk = 0; kk < DIMF; kk += 4) {
            const int ka = kk + 2 * half;
            v2f a, b;
            a.x = sA[l16][ka];
            a.y = sA[l16][ka + 1];
            b.x = W1[(size_t)ka * DIMF + ct + l16];
            b.y = W1[(size_t)(ka + 1) * DIMF + ct + l16];
            c = __builtin_amdgcn_wmma_f32_16x16x4_f32(false, a, false, b,
                                                      (short)0, c, false, false);
        }
        const float bias = b1[ct + l16];
        for (int r = 0; r < 8; ++r) {
            float v = c[r] + bias;
            sB[r + 8 * half][ct + l16] = v > 0.0f ? v : 0.0f;
        }
    }
    __syncthreads();

    // ---- GEMM2 + bias + relu -> global, with BN partial stats ----
    for (int t = 0; t < 2; ++t) {
        const int ct = wave * 32 + t * 16;
        v8f c = {};
        for (int kk = 0; kk < DIMF; kk += 4) {
            const int ka = kk + 2 * half;
            v2f a, b;
            a.x = sB[l16][ka];
            a.y = sB[l16][ka + 1];
            b.x = W2[(size_t)ka * DIMF + ct + l16];
            b.y = W2[(size_t)(ka + 1) * DIMF + ct + l16];
            c = __builtin_amdgcn_wmma_f32_16x16x4_f32(false, a, false, b,
                                                      (short)0, c, false, false);
        }
        const float bias = b2[ct + l16];
        float s = 0.0f, s2 = 0.0f;
        for (int r = 0; r < 8; ++r) {
            float v = c[r] + bias;
            v = v > 0.0f ? v : 0.0f;
            hout[(row0 + r + 8 * half) * DIMF + ct + l16] = v;
            s  += v;
            s2 += v * v;
        }
        // lanes l and l^16 hold the same feature -> pairwise combine, then atomic
        s  += __shfl_xor(s, 16, 32);
        s2 += __shfl_xor(s2, 16, 32);
        if (half == 0) {
            atomicAdd(&bn_sum[ct + l16], s);
            atomicAdd(&bn_sumsq[ct + l16], s2);
        }
    }
}

// ------------------------------ BN finalize --------------------------------
__global__ void bn_finalize_kernel(const float* __restrict__ sum,
                                   const float* __restrict__ sumsq,
                                   const float* __restrict__ gamma,
                                   const float* __restrict__ beta,
                                   float* __restrict__ scale,
                                   float* __restrict__ shift, float invN)
{
    const int f = threadIdx.x;
    const float mean = sum[f] * invN;
    const float var  = sumsq[f] * invN - mean * mean;
    const float sc   = gamma[f] * rsqrtf(var + 1e-5f);
    scale[f] = sc;
    shift[f] = beta[f] - mean * sc;
}

// ------------------------- codebook normalization --------------------------
// One block per (res-layer, code) row: cbn = cb * rsqrt(sum cb^2 + 1e-12)
__global__ __launch_bounds__(128) void cbn_kernel(const float* __restrict__ cb,
                                                  float* __restrict__ cbn)
{
    __shared__ float red[128];
    const int row = blockIdx.x;          // 0..47 = l*16 + code
    const int f   = threadIdx.x;
    const float v = cb[(size_t)row * DIMF + f];
    red[f] = v * v;
    __syncthreads();
    for (int s = 64; s > 0; s >>= 1) {
        if (f < s) red[f] += red[f + s];
        __syncthreads();
    }
    cbn[(size_t)row * DIMF + f] = v * rsqrtf(red[0] + 1e-12f);
}

// -------- fused BN-apply + xs_cat write + graph pool + residual VQ ---------
// One wave per node. argmax(l2norm(r)·cbn) == argmax(r·cbn) since the row
// norm is a positive scalar, so the per-node normalization is elided.
__global__ __launch_bounds__(256) void bn_vq_pool_kernel(
    float* __restrict__ h,               // [N,128] in: pre-BN, out: post-BN
    const float* __restrict__ scale, const float* __restrict__ shift,
    const float* __restrict__ cbn,       // [3*16,128] normalized codebooks
    const int* __restrict__ batch,
    float* __restrict__ xs_out,          // [N,384]
    float* __restrict__ xpool,           // [512,384]
    float* __restrict__ id_out,          // [N,9] as float
    float* __restrict__ loss_acc,
    int layer, int nNodes)
{
    const int node = blockIdx.x * 8 + (threadIdx.x >> 5);
    if (node >= nNodes) return;
    const int lane = threadIdx.x & 31;
    const int f0 = lane * 4;

    const float4 hv = *(const float4*)(h + (size_t)node * DIMF + f0);
    const float4 sc = *(const float4*)(scale + f0);
    const float4 sh = *(const float4*)(shift + f0);
    float r[4];
    r[0] = hv.x * sc.x + sh.x;
    r[1] = hv.y * sc.y + sh.y;
    r[2] = hv.z * sc.z + sh.z;
    r[3] = hv.w * sc.w + sh.w;

    float4 o; o.x = r[0]; o.y = r[1]; o.z = r[2]; o.w = r[3];
    *(float4*)(h + (size_t)node * DIMF + f0) = o;                       // h for next layer
    *(float4*)(xs_out + (size_t)node * 384 + layer * DIMF + f0) = o;    // xs_cat slice

    const int g = batch[node];
    float* xp = xpool + (size_t)g * 384 + layer * DIMF + f0;
    atomicAdd(xp + 0, r[0]);
    atomicAdd(xp + 1, r[1]);
    atomicAdd(xp + 2, r[2]);
    atomicAdd(xp + 3, r[3]);

    float sse = 0.0f;
    for (int l = 0; l < NRES; ++l) {
        const float* cb = cbn + (size_t)l * NCODE * DIMF;
        float best = -3.4e38f;
        int bidx = 0;
        for (int cI = 0; cI < NCODE; ++cI) {
            const float* cr = cb + (size_t)cI * DIMF + f0;
            float p = r[0] * cr[0] + r[1] * cr[1] + r[2] * cr[2] + r[3] * cr[3];
            p += __shfl_xor(p, 1, 32);
            p += __shfl_xor(p, 2, 32);
            p += __shfl_xor(p, 4, 32);
            p += __shfl_xor(p, 8, 32);
            p += __shfl_xor(p, 16, 32);
            if (p > best) { best = p; bidx = cI; }   // strict > == first-max tie-break
        }
        const float* q = cb + (size_t)bidx * DIMF + f0;
        for (int j = 0; j < 4; ++j) {
            const float d = q[j] - r[j];
            sse += d * d;
            r[j] -= q[j];
        }
        if (lane == 0) id_out[(size_t)node * 9 + layer * 3 + l] = (float)bidx;
    }
    sse += __shfl_xor(sse, 1, 32);
    sse += __shfl_xor(sse, 2, 32);
    sse += __shfl_xor(sse, 4, 32);
    sse += __shfl_xor(sse, 8, 32);
    sse += __shfl_xor(sse, 16, 32);
    if (lane == 0) atomicAdd(loss_acc, sse);
}

__global__ void loss_final_kernel(const float* __restrict__ acc, float* __restrict__ out) {
    out[0] = acc[0] * (0.25f / ((float)NNODES * (float)DIMF));
}

// ------------------------------- launcher ----------------------------------

extern "C" void kernel_launch(void* const* d_in, const int* in_sizes, int n_in,
                              void* d_out, int out_size, void* d_ws, size_t ws_size,
                              hipStream_t stream) {
    (void)in_sizes; (void)n_in; (void)out_size; (void)ws_size;

    const float* x     = (const float*)d_in[0];  // [N,128]
    const int*   ei    = (const int*)d_in[1];    // [2,E]
    const int*   batch = (const int*)d_in[2];    // [N]
    const float* W1    = (const float*)d_in[3];  // [3,128,128]
    const float* b1    = (const float*)d_in[4];  // [3,128]
    const float* W2    = (const float*)d_in[5];
    const float* b2    = (const float*)d_in[6];
    const float* gamma = (const float*)d_in[7];
    const float* beta  = (const float*)d_in[8];
    const float* cbs   = (const float*)d_in[9];  // [3,3,16,128]

    float* out       = (float*)d_out;
    float* out_xpool = out;                                   // 512*384
    float* out_xs    = out + (size_t)NGRAPHS * 384;           // N*384
    float* out_loss  = out_xs + (size_t)NNODES * 384;         // 1
    float* out_id    = out_loss + 1;                          // N*9

    float* ws   = (float*)d_ws;
    float* h    = ws;                                         // N*128
    float* agg  = h + (size_t)NNODES * DIMF;                  // N*128
    float* bn   = agg + (size_t)NNODES * DIMF;                // sum|sumsq|scale|shift = 512
    float* cbn  = bn + 512;                                   // 3*16*128
    float* loss = cbn + NRES * NCODE * DIMF;                  // 1

    const int ND4 = NNODES * DIMF / 4;

    zero_kernel<<<divUp(NGRAPHS * 384, 256), 256, 0, stream>>>(out_xpool, NGRAPHS * 384);
    zero_kernel<<<1, 32, 0, stream>>>(loss, 1);
    copy4_kernel<<<divUp(ND4, 256), 256, 0, stream>>>((const float4*)x, (float4*)h, ND4);

    for (int i = 0; i < NLAYERS; ++i) {
        // agg = h + segment_sum(h[src] over dst)
        copy4_kernel<<<divUp(ND4, 256), 256, 0, stream>>>((const float4*)h, (float4*)agg, ND4);
        scatter_add_kernel<<<NEDGES / 8, 256, 0, stream>>>(h, ei, ei + NEDGES, agg, NEDGES);

        zero_kernel<<<1, 256, 0, stream>>>(bn, 256);
        gin_mlp_kernel<<<NNODES / 16, 128, 0, stream>>>(
            agg, W1 + (size_t)i * DIMF * DIMF, b1 + i * DIMF,
            W2 + (size_t)i * DIMF * DIMF, b2 + i * DIMF, h, bn, bn + 128);

        bn_finalize_kernel<<<1, 128, 0, stream>>>(bn, bn + 128, gamma + i * DIMF,
                                                  beta + i * DIMF, bn + 256, bn + 384,
                                                  1.0f / (float)NNODES);
        cbn_kernel<<<NRES * NCODE, 128, 0, stream>>>(cbs + (size_t)i * NRES * NCODE * DIMF, cbn);

        bn_vq_pool_kernel<<<divUp(NNODES, 8), 256, 0, stream>>>(
            h, bn + 256, bn + 384, cbn, batch, out_xs, out_xpool, out_id, loss, i, NNODES);
    }

    loss_final_kernel<<<1, 1, 0, stream>>>(loss, out_loss);
}